// CompressedActivation_69380901700186
// MI455X (gfx1250) — compile-verified
//
#include <hip/hip_runtime.h>
#include <hip/hip_bf16.h>

// CompressedActivation forward == identity (compress->decompress round trip).
// Pure streaming copy: 256 MiB in + 256 MiB out, HBM-bound (~23 us @ 23.3 TB/s).
// Zero FLOPs -> WMMA is inapplicable; the CDNA5-specific win is the data path:
//   * NT temporal hints (512 MiB working set > 192 MB L2, zero reuse)
//   * gfx1250 async global<->LDS streaming ops (ASYNCcnt / s_wait_asynccnt)
//   * wave32-sized tiling (32 lanes x 16 B = 512 B per async instruction)

#define THREADS        256
#define WAVES_PER_BLK  8            // 256 / wave32
#define CHUNK_BYTES    512          // 32 lanes * 16 B per async b128 instruction
#define CHUNKS         8            // chunks per tile
#define TILE_BYTES     (CHUNKS * CHUNK_BYTES)   // 4 KiB per wave per tile

// native 16-byte vector type (HIP's float4 is a class and is rejected by
// __builtin_nontemporal_load/store)
typedef float v4f __attribute__((ext_vector_type(4)));

// ---------------------------------------------------------------------------
// Path A: async-LDS streaming copy (gfx1250 GLOBAL_*_ASYNC_*_LDS_B128).
// Each wave moves 4 KiB tiles HBM -> LDS -> HBM, always via the same one of its two LDS buffers.
// LDS reuse: tile t writes buffer (t&1); t steps by nwaves, a multiple of 8, so
// every tile of a wave reuses that buffer, and its loads are issued before the
// s_wait_asynccnt 0 that drains the previous tile's async stores.
// ---------------------------------------------------------------------------
__global__ void __launch_bounds__(THREADS)
copy_async_lds_kernel(const unsigned char* __restrict__ src,
                      unsigned char* __restrict__ dst,
                      long long tiles)
{
    // 8 waves * 2 buffers * 4 KiB = 64 KiB (of 320 KiB/WGP)
    __shared__ __align__(16) unsigned char ldsbuf[WAVES_PER_BLK * 2 * TILE_BYTES];

    const int lane  = threadIdx.x & 31;
    const int wave  = threadIdx.x >> 5;
    const long long gwave  = (long long)blockIdx.x * WAVES_PER_BLK + wave;
    const long long nwaves = (long long)gridDim.x * WAVES_PER_BLK;

    for (long long t = gwave; t < tiles; t += nwaves) {
        const long long base = t * (long long)TILE_BYTES;
        const unsigned bufSel = (unsigned)(t & 1);
        unsigned char* wbuf = &ldsbuf[(wave * 2 + bufSel) * TILE_BYTES];

        // Issue CHUNKS async loads: HBM -> LDS (ASYNCcnt += CHUNKS)
        #pragma unroll
        for (int u = 0; u < CHUNKS; ++u) {
            // generic (flat) LDS pointer: bits [31:0] are the LDS byte offset
            unsigned lds_addr = (unsigned)(unsigned long long)
                (void*)&wbuf[u * CHUNK_BYTES + lane * 16];
            const unsigned char* gaddr = src + base + u * CHUNK_BYTES + lane * 16;
            asm volatile("global_load_async_to_lds_b128 %0, %1, off"
                         :: "v"(lds_addr), "v"(gaddr)
                         : "memory");
        }
        // Wait until the tile is resident in LDS (also drains prior stores)
        asm volatile("s_wait_asynccnt 0x0" ::: "memory");

        // Issue CHUNKS async stores: LDS -> HBM (completion drained next iter)
        #pragma unroll
        for (int u = 0; u < CHUNKS; ++u) {
            unsigned lds_addr = (unsigned)(unsigned long long)
                (void*)&wbuf[u * CHUNK_BYTES + lane * 16];
            unsigned char* gaddr = dst + base + u * CHUNK_BYTES + lane * 16;
            asm volatile("global_store_async_from_lds_b128 %0, %1, off"
                         :: "v"(gaddr), "v"(lds_addr)
                         : "memory");
        }
    }
    // Drain all outstanding async stores before wave exit
    asm volatile("s_wait_asynccnt 0x0" ::: "memory");
}

// ---------------------------------------------------------------------------
// Path B: direct non-temporal b128 grid-stride copy over floats [start, n).
// global_load_b128 / global_store_b128 with TH=NT (no reuse -> bypass L2
// retention). Scalar NT tail for n % 4 (not hit for 8192^2 but kept generic).
// ---------------------------------------------------------------------------
__global__ void __launch_bounds__(THREADS)
copy_nt_kernel(const float* __restrict__ src, float* __restrict__ dst,
               long long start, long long n)
{
    const long long tid    = (long long)blockIdx.x * blockDim.x + threadIdx.x;
    const long long stride = (long long)gridDim.x * blockDim.x;

    const long long m  = n - start;
    const long long m4 = m >> 2;
    const v4f* __restrict__ s4 = (const v4f*)(src + start);
    v4f* __restrict__ d4       = (v4f*)(dst + start);

    for (long long i = tid; i < m4; i += stride) {
        v4f v = __builtin_nontemporal_load(s4 + i);
        __builtin_nontemporal_store(v, d4 + i);
    }
    // scalar tail
    for (long long i = (m4 << 2) + tid; i < m; i += stride) {
        float f = __builtin_nontemporal_load(src + start + i);
        __builtin_nontemporal_store(f, dst + start + i);
    }
}

extern "C" void kernel_launch(void* const* d_in, const int* in_sizes, int n_in,
                              void* d_out, int out_size, void* d_ws, size_t ws_size,
                              hipStream_t stream) {
    (void)in_sizes; (void)n_in; (void)d_ws; (void)ws_size;

    const float* x   = (const float*)d_in[0];
    float*       out = (float*)d_out;
    const long long n      = (long long)out_size;      // 8192*8192 floats
    const long long nBytes = n * 4;

    // Front 1/8 of the buffer (4 KiB-tile aligned) via the async-LDS path.
    long long asyncBytes = (nBytes / 8) & ~((long long)TILE_BYTES - 1);
    if (asyncBytes < 0) asyncBytes = 0;
    const long long tiles  = asyncBytes / TILE_BYTES;
    const long long asyncN = asyncBytes / 4;           // floats covered by path A

    if (tiles > 0) {
        // ~4 tiles (16 KiB) per wave
        long long blocks = (tiles + (long long)WAVES_PER_BLK * 4 - 1)
                           / ((long long)WAVES_PER_BLK * 4);
        if (blocks < 1) blocks = 1;
        if (blocks > 1048576) blocks = 1048576;
        copy_async_lds_kernel<<<(unsigned)blocks, THREADS, 0, stream>>>(
            (const unsigned char*)x, (unsigned char*)out, tiles);
    }

    if (n > asyncN) {
        // ~8 float4 (128 B) per thread: enough concurrency to saturate HBM
        long long m4 = (n - asyncN) >> 2;
        long long blocks = (m4 + (long long)THREADS * 8 - 1)
                           / ((long long)THREADS * 8);
        if (blocks < 1) blocks = 1;
        if (blocks > 1048576) blocks = 1048576;
        copy_nt_kernel<<<(unsigned)blocks, THREADS, 0, stream>>>(x, out, asyncN, n);
    }
}